// MultiAxisAttention_21122649162485
// MI455X (gfx1250) — compile-verified
//
#include <hip/hip_runtime.h>

// MultiAxisAttention for MI455X (gfx1250, wave32).
// fp32 end-to-end using V_WMMA_F32_16X16X4_F32, async global->LDS staging
// (GLOBAL_LOAD_ASYNC_TO_LDS_B128 / ASYNCcnt), and a double-buffered,
// ASYNCcnt-pipelined K loop in the projection GEMM.
//
// Phase 1: qkv = X(32768x512) @ W(512x1536)  -> d_ws
// Phase 2: per (b, head, axis-index): 64x64 attention with softmax -> d_out

typedef __attribute__((ext_vector_type(2))) float v2f;
typedef __attribute__((ext_vector_type(4))) float v4f;
typedef __attribute__((ext_vector_type(8))) float v8f;
typedef int gv4i __attribute__((vector_size(16)));   // matches builtin param type

namespace cfg {
constexpr int B = 8, N = 64, P = 64, C = 512, H = 8, D = 64;
constexpr int M1 = B * N * P;   // 32768 GEMM rows
constexpr int K1 = C;           // 512
constexpr int N1 = 3 * C;       // 1536
constexpr int LS  = 68;  // attention LDS row stride (floats): 68%64=4, 16B aligned
constexpr int XST = 36;  // gemm X-tile row stride (32 cols + 4 pad): conflict-free, 16B aligned
constexpr int WST = 68;  // gemm W-tile row stride (64 cols + 4 pad)
}

#if __has_builtin(__builtin_amdgcn_global_load_async_to_lds_b128)
#define USE_ASYNC_LDS 1
#else
#define USE_ASYNC_LDS 0
#endif

#if USE_ASYNC_LDS
#if __has_builtin(__builtin_amdgcn_s_wait_asynccnt)
#define ASYNC_WAIT(n) __builtin_amdgcn_s_wait_asynccnt(n)
#else
#define ASYNC_WAIT(n) asm volatile("s_wait_asynccnt %0" ::"i"(n) : "memory")
#endif
#else
#define ASYNC_WAIT(n) ((void)0)
#endif

// Per-lane 16-byte global -> LDS transfer. Async path bypasses VGPRs and is
// tracked with ASYNCcnt (in-order completion); fallback is load+ds_store.
__device__ __forceinline__ void stage16(const float* __restrict__ gsrc, float* ldst) {
#if USE_ASYNC_LDS
    __builtin_amdgcn_global_load_async_to_lds_b128(
        (__attribute__((address_space(1))) gv4i*)(void*)gsrc,
        (__attribute__((address_space(3))) gv4i*)(void*)ldst,
        /*offset=*/0, /*cpol=*/0);
#else
    *(v4f*)ldst = *(const v4f*)gsrc;
#endif
}

// ---------------------------------------------------------------------------
// Kernel 1: QKV projection GEMM. Block computes a 64x64 tile of qkv.
// 4 waves; wave w owns rows [16w,16w+16), all 64 cols (4 accumulators).
// K streamed in 32-chunks, double-buffered: chunk k+1's async transfers are
// in flight while chunk k's WMMAs execute; wait ASYNCcnt<=8 (next chunk's
// in-flight ops) instead of 0.
// ---------------------------------------------------------------------------
__global__ __launch_bounds__(128) void qkv_gemm_kernel(const float* __restrict__ X,
                                                       const float* __restrict__ W,
                                                       float* __restrict__ QKV) {
    using namespace cfg;
    __shared__ float xs[2][64 * XST];   // X tile: [m][k], k-chunk of 32
    __shared__ float ws[2][32 * WST];   // W tile: [k][n]

    const int tid  = threadIdx.x;
    const int wave = tid >> 5;
    const int lane = tid & 31;
    const int mbase = blockIdx.x * 64;
    const int nbase = blockIdx.y * 64;

    // Issue one chunk's staging: 4 X transfers + 4 W transfers per thread.
    auto issue_chunk = [&](int kc, int buf) {
        const int kb = kc * 32;
        for (int i = tid; i < 64 * 8; i += 128) {       // X: 64 rows x 8 float4
            const int r  = i >> 3;
            const int c4 = (i & 7) * 4;
            stage16(X + (size_t)(mbase + r) * K1 + kb + c4, &xs[buf][r * XST + c4]);
        }
        for (int i = tid; i < 32 * 16; i += 128) {      // W: 32 rows x 16 float4
            const int r  = i >> 4;
            const int c4 = (i & 15) * 4;
            stage16(W + (size_t)(kb + r) * N1 + nbase + c4, &ws[buf][r * WST + c4]);
        }
    };

    v8f acc[4] = {};

    const int mrow   = wave * 16 + (lane & 15);   // A-frag row (LDS-local)
    const int khalf  = (lane >> 4) * 2;           // lanes 16-31 take K = +2,+3
    const int ncol16 = lane & 15;                 // B/C column within 16-tile

    issue_chunk(0, 0);

    for (int kc = 0; kc < K1 / 32; ++kc) {
        const int cur = kc & 1;
        if (kc + 1 < K1 / 32) {
            issue_chunk(kc + 1, cur ^ 1);   // overlaps with this chunk's compute
            ASYNC_WAIT(8);                  // current chunk done; next still in flight
        } else {
            ASYNC_WAIT(0);
        }
        __syncthreads();

        const float* xb = xs[cur];
        const float* wb = ws[cur];
        // 8 K-steps of 4; A-frag reused across the 4 N-tiles.
        for (int ksi = 0; ksi < 8; ++ksi) {
            const int k0 = ksi * 4 + khalf;
            const v2f a = *(const v2f*)(xb + mrow * XST + k0);
#pragma unroll
            for (int nt = 0; nt < 4; ++nt) {
                v2f b;
                b.x = wb[(k0 + 0) * WST + nt * 16 + ncol16];
                b.y = wb[(k0 + 1) * WST + nt * 16 + ncol16];
                acc[nt] = __builtin_amdgcn_wmma_f32_16x16x4_f32(
                    false, a, false, b, (short)0, acc[nt], false, false);
            }
        }
        __syncthreads();   // all waves done with buf[cur] before it is re-staged
    }

    // Store: C/D layout: m = reg + 8*(lane>=16), n = lane&15.
    const int mhi = (lane >> 4) * 8;
#pragma unroll
    for (int nt = 0; nt < 4; ++nt) {
#pragma unroll
        for (int r = 0; r < 8; ++r) {
            const int m = mbase + wave * 16 + mhi + r;
            const int n = nbase + nt * 16 + ncol16;
            QKV[(size_t)m * N1 + n] = acc[nt][r];
        }
    }
}

// ---------------------------------------------------------------------------
// Kernel 2: attention per (b, head, axis index a). 4096 blocks, 128 threads.
// Heads 0..3 attend along n (row stride 64 rows of qkv); heads 4..7 along p.
// ---------------------------------------------------------------------------
__global__ __launch_bounds__(128) void attention_kernel(const float* __restrict__ QKV,
                                                        float* __restrict__ Out) {
    using namespace cfg;
    __shared__ float qs[64 * LS];   // q tile; later overlaid with sim/att
    __shared__ float kt[64 * LS];   // k tile
    __shared__ float vt[64 * LS];   // v tile

    const int tid  = threadIdx.x;
    const int wave = tid >> 5;
    const int lane = tid & 31;

    const int bid = blockIdx.x;
    const int b   = bid >> 9;         // /512
    const int h   = (bid >> 6) & 7;
    const int a   = bid & 63;
    const bool swapped = (h < 4);     // axis-split: first half of heads swap n<->p

    const int qc = 0 * C + h * D;
    const int kc = 1 * C + h * D;
    const int vc = 2 * C + h * D;

    // Stage q, k, v: 64 rows x 64 floats each (async direct-to-LDS).
    for (int i = tid; i < 64 * 16; i += 128) {
        const int s  = i >> 4;
        const int c4 = (i & 15) * 4;
        const size_t row = (size_t)b * 4096 + (swapped ? (s * 64 + a) : (a * 64 + s));
        const float* src = QKV + row * (size_t)N1;
        stage16(src + qc + c4, qs + s * LS + c4);
        stage16(src + kc + c4, kt + s * LS + c4);
        stage16(src + vc + c4, vt + s * LS + c4);
    }
    ASYNC_WAIT(0);
    __syncthreads();

    const int mrow   = wave * 16 + (lane & 15);
    const int khalf  = (lane >> 4) * 2;
    const int ncol16 = lane & 15;
    const int mhi    = (lane >> 4) * 8;

    // sim = q @ k^T   (A[m=s][k=dd] = qs, B[k=dd][n=s2] = kt[s2][dd])
    v8f sim[4] = {};
    for (int ksi = 0; ksi < 16; ++ksi) {
        const int k0 = ksi * 4 + khalf;
        const v2f afrag = *(const v2f*)(qs + mrow * LS + k0);
#pragma unroll
        for (int nt = 0; nt < 4; ++nt) {
            const int n = nt * 16 + ncol16;
            const v2f bfrag = *(const v2f*)(kt + n * LS + k0);  // contiguous dd pair
            sim[nt] = __builtin_amdgcn_wmma_f32_16x16x4_f32(
                false, afrag, false, bfrag, (short)0, sim[nt], false, false);
        }
    }

    // Apply attention scale post-GEMM: scale*(q.k) == (scale*q).k
#pragma unroll
    for (int nt = 0; nt < 4; ++nt) sim[nt] *= 0.125f;   // (C/H)^-0.5 = 64^-0.5

    // Overlay sim onto qs: wave w only reads/writes its own 16 rows -> safe.
#pragma unroll
    for (int nt = 0; nt < 4; ++nt) {
#pragma unroll
        for (int r = 0; r < 8; ++r) {
            qs[(wave * 16 + mhi + r) * LS + nt * 16 + ncol16] = sim[nt][r];
        }
    }
    __syncthreads();

    // Row softmax (max-subtracted) on qs rows; 64 threads, one row each.
    if (tid < 64) {
        float* rowp = qs + tid * LS;
        float mx = rowp[0];
        for (int j = 1; j < 64; ++j) mx = fmaxf(mx, rowp[j]);
        float sum = 0.0f;
        for (int j = 0; j < 64; ++j) {
            const float e = __expf(rowp[j] - mx);
            rowp[j] = e;
            sum += e;
        }
        const float inv = 1.0f / sum;
        for (int j = 0; j < 64; ++j) rowp[j] *= inv;
    }
    __syncthreads();

    // y = att @ v   (A[m=s][k=s2] = qs, B[k=s2][n=dd] = vt[s2][dd])
    v8f yacc[4] = {};
    for (int ksi = 0; ksi < 16; ++ksi) {
        const int k0 = ksi * 4 + khalf;
        const v2f afrag = *(const v2f*)(qs + mrow * LS + k0);
#pragma unroll
        for (int nt = 0; nt < 4; ++nt) {
            const int n = nt * 16 + ncol16;
            v2f bfrag;
            bfrag.x = vt[(k0 + 0) * LS + n];
            bfrag.y = vt[(k0 + 1) * LS + n];
            yacc[nt] = __builtin_amdgcn_wmma_f32_16x16x4_f32(
                false, afrag, false, bfrag, (short)0, yacc[nt], false, false);
        }
    }

    // Faithful output: out[b, a, s, h*64+dd] = y'[s][dd] (swap NOT undone).
    float* outBase = Out + ((size_t)b * 4096 + (size_t)a * 64) * C + h * D;
#pragma unroll
    for (int nt = 0; nt < 4; ++nt) {
#pragma unroll
        for (int r = 0; r < 8; ++r) {
            const int s  = wave * 16 + mhi + r;
            const int dd = nt * 16 + ncol16;
            outBase[(size_t)s * C + dd] = yacc[nt][r];
        }
    }
}

// ---------------------------------------------------------------------------
extern "C" void kernel_launch(void* const* d_in, const int* in_sizes, int n_in,
                              void* d_out, int out_size, void* d_ws, size_t ws_size,
                              hipStream_t stream) {
    using namespace cfg;
    (void)in_sizes; (void)n_in; (void)out_size; (void)ws_size;
    const float* X = (const float*)d_in[0];   // (B,N,P,C) = 32768 x 512
    const float* W = (const float*)d_in[1];   // (C, 3C)   = 512 x 1536
    float* qkv = (float*)d_ws;                // 32768 x 1536 fp32 scratch
    float* out = (float*)d_out;               // (B,N,P,C)

    dim3 g1(M1 / 64, N1 / 64);                // 512 x 24
    qkv_gemm_kernel<<<g1, 128, 0, stream>>>(X, W, qkv);
    attention_kernel<<<B * H * 64, 128, 0, stream>>>(qkv, out);
}